// MyModel_61933428410700
// MI455X (gfx1250) — compile-verified
//
#include <hip/hip_runtime.h>
#include <hip/hip_bf16.h>
#include <stdint.h>

// ---------------- problem constants (match reference) ----------------
#define BLANK   14
#define NEGF    (-1e30f)
#define S_CAP   200
#define L_CAP   (2 * S_CAP + 1)     // 401
#define C_CAP   256
#define TBLK    416                 // 13 wave32 waves; covers L=401 lanes

// ---------------- CDNA5 async global->LDS helpers ----------------
typedef int v4i __attribute__((ext_vector_type(4)));
typedef __attribute__((address_space(1))) v4i   as1_v4i;
typedef __attribute__((address_space(3))) v4i   as3_v4i;
typedef __attribute__((address_space(3))) float as3_float;

__device__ __forceinline__ void async_load_b128(const float* g, float* l) {
#if __has_builtin(__builtin_amdgcn_global_load_async_to_lds_b128)
    __builtin_amdgcn_global_load_async_to_lds_b128(
        (as1_v4i*)(const_cast<float*>(g)), (as3_v4i*)l, 0, 0);
#else
    unsigned lofs = (unsigned)(size_t)(as3_float*)l;      // 32-bit LDS offset
    unsigned long long ga = (unsigned long long)(size_t)g;
    asm volatile("global_load_async_to_lds_b128 %0, %1, off"
                 :: "v"(lofs), "v"(ga) : "memory");
#endif
}

__device__ __forceinline__ void wait_async_le1() {
#if __has_builtin(__builtin_amdgcn_s_wait_asynccnt)
    __builtin_amdgcn_s_wait_asynccnt(1);
#else
    asm volatile("s_wait_asynccnt 0x1" ::: "memory");
#endif
}

__device__ __forceinline__ void wait_async_0() {
#if __has_builtin(__builtin_amdgcn_s_wait_asynccnt)
    __builtin_amdgcn_s_wait_asynccnt(0);
#else
    asm volatile("s_wait_asynccnt 0x0" ::: "memory");
#endif
}

// ---------------- CTC alpha recursion: one workgroup per batch b ----------------
__global__ __launch_bounds__(TBLK)
void ctc_alpha_kernel(const float* __restrict__ lp,    // (T,B,C) log-probs
                      const int*   __restrict__ tgt,   // (B,S)
                      const int*   __restrict__ ilen,  // (B,)
                      const int*   __restrict__ tlen,  // (B,)
                      float*       __restrict__ out,   // scalar accumulator
                      int T, int B, int C, int S) {
    __shared__ float rowbuf[3][C_CAP];        // 3-deep async pipeline of emission rows
    __shared__ float abuf[2][L_CAP + 2];      // double-buffered alpha, 2-slot NEG pad

    const int  b      = blockIdx.x;
    const int  tid    = threadIdx.x;
    const int  L      = 2 * S + 1;
    const bool lat    = (tid < L);            // lattice lane
    const bool loader = (tid < 64);           // async-copy lanes (waves 0,1)

    // per-lane extended label z[s] and skip-transition flag
    int  z    = BLANK;
    bool skip = false;
    if (lat && (tid & 1)) {
        z = tgt[(size_t)b * S + (tid >> 1)];
        if (tid >= 2)
            skip = (z != tgt[(size_t)b * S + (tid >> 1) - 1]);
    }

    const size_t rowstride = (size_t)B * C;            // floats between time steps
    const float* gbase     = lp + (size_t)b * C;       // row t at gbase + t*rowstride
    const int    lo        = (tid & 63) * 4;           // loader lane LDS/global lane offset

    // prime the pipeline: rows 0 and 1 in flight (64 lanes x 16B = 1KB row)
    if (loader) {
        async_load_b128(gbase + lo,             &rowbuf[0][lo]);
        if (T > 1)
            async_load_b128(gbase + rowstride + lo, &rowbuf[1][lo]);
    }
    wait_async_le1();          // row 0 complete (row 1 may still be in flight)
    __syncthreads();

    // alpha_0
    float alpha = NEGF;
    if (tid < 2 && lat) alpha = rowbuf[0][z];
    if (lat)     abuf[0][tid + 2] = alpha;
    if (tid < 2) { abuf[0][tid] = NEGF; abuf[1][tid] = NEGF; }
    const int IL = ilen[b];

    // rotating row-buffer pointers: at iteration t, rR holds row t (t%3),
    // dD is the async destination for row t+1 ((t+1)%3)
    const float* rR = rowbuf[1];
    const float* rS = rowbuf[2];
    const float* rQ = rowbuf[0];
    float* dD = &rowbuf[2][lo];
    float* dE = &rowbuf[0][lo];
    float* dF = &rowbuf[1][lo];
    const float* gp = gbase + 2 * rowstride + lo;      // row t+1 global addr, lane part

    const float* aR = abuf[0];     // read alpha buffer
    float*       aW = abuf[1];     // write alpha buffer

    #pragma unroll 3
    for (int t = 1; t < T; ++t) {
        // prefetch row t+1 into the buffer last read two barriers ago
        if ((t + 1) < T) {
            if (loader) async_load_b128(gp, dD);
            wait_async_le1();      // row t done; row t+1 outstanding
        } else {
            wait_async_0();        // last step: drain so row T-1 is resident
        }
        __syncthreads();           // publish alpha[t-1] + emission row t to all waves

        if (lat) {
            float a0 = aR[tid + 2];
            float a1 = aR[tid + 1];
            float a2 = skip ? aR[tid] : NEGF;
            // 3-way logaddexp (max + log(sum exp))
            float m  = fmaxf(a0, fmaxf(a1, a2));
            float pv = m + __logf(__expf(a0 - m) + __expf(a1 - m) + __expf(a2 - m));
            float na = pv + rR[z];
            if (t < IL) alpha = na;            // freeze past input_length
            aW[tid + 2] = alpha;
        }
        // rotate buffers / advance addresses (register moves only)
        { const float* tr = rR; rR = rS; rS = rQ; rQ = tr; }
        { float* td = dD; dD = dE; dE = dF; dF = td; }
        { const float* ta = aR; aR = aW; aW = (float*)ta; }
        gp += rowstride;
    }
    __syncthreads();

    if (tid == 0) {
        int   tl = tlen[b];
        float l1 = aR[2 * tl + 2];             // alpha[2*tl]   (final blank)
        float l2 = aR[2 * tl + 1];             // alpha[2*tl-1] (final label)
        float m  = fmaxf(l1, l2);
        float nll = -(m + __logf(__expf(l1 - m) + __expf(l2 - m)));
        atomicAdd(out, nll / ((float)tl * (float)B));
    }
}

// ---------------- host-side launcher ----------------
extern "C" void kernel_launch(void* const* d_in, const int* in_sizes, int n_in,
                              void* d_out, int out_size, void* d_ws, size_t ws_size,
                              hipStream_t stream) {
    const float* lp   = (const float*)d_in[0];
    const int*   tgt  = (const int*)d_in[1];
    const int*   ilen = (const int*)d_in[2];
    const int*   tlen = (const int*)d_in[3];
    float*       out  = (float*)d_out;

    const int B = in_sizes[2];                 // 64
    const int S = in_sizes[1] / B;             // 200
    const int C = C_CAP;                       // 256
    const int T = (int)(in_sizes[0] / ((long long)B * C));  // 2000

    (void)hipMemsetAsync(out, 0, (size_t)out_size * sizeof(float), stream);
    ctc_alpha_kernel<<<B, TBLK, 0, stream>>>(lp, tgt, ilen, tlen, out, T, B, C, S);
}